// NMLinear_13623636263495
// MI455X (gfx1250) — compile-verified
//
#include <hip/hip_runtime.h>

typedef __attribute__((ext_vector_type(2))) float v2f;
typedef __attribute__((ext_vector_type(8))) float v8f;

constexpr int TOKENS = 64;
constexpr int IN_F   = 8192;
constexpr int OUT_F  = 8192;
constexpr int KSPLIT = 16;                 // K-dimension split for parallelism
constexpr int KCHUNK = IN_F / KSPLIT;      // 512 -> 128 k-steps per wave
constexpr int WAVES_PER_BLOCK = 8;         // 256 threads, wave32
constexpr int NTILES = OUT_F / 16;         // 512

// ---------------------------------------------------------------------------
// Kernel 1: out[m][n] = bias[n]  (epilogue base for atomic accumulation)
// ---------------------------------------------------------------------------
__global__ void nm_bias_init(const float* __restrict__ bias,
                             float* __restrict__ out) {
    int i = blockIdx.x * blockDim.x + threadIdx.x;   // 64*8192 total
    out[i] = bias[i & (OUT_F - 1)];
}

// ---------------------------------------------------------------------------
// Kernel 2: streamed masked GEMM via v_wmma_f32_16x16x4_f32.
// Wave = one 16-col N-tile x one K-chunk; all 4 M-tiles (64 tokens) live in
// 32 accumulator VGPRs so weight+logits stream exactly once from DRAM.
// ---------------------------------------------------------------------------
__global__ __launch_bounds__(WAVES_PER_BLOCK * 32)
void nm_linear_wmma(const float* __restrict__ x,
                    const float* __restrict__ w,
                    const float* __restrict__ logits,
                    float* __restrict__ out) {
    const int lane = threadIdx.x & 31;
    const int wave = threadIdx.x >> 5;
    const int half = lane >> 4;   // 0: K=k,k+1 ; 1: K=k+2,k+3 (A & B frag layout)
    const int l16  = lane & 15;

    const int ntile = blockIdx.x * WAVES_PER_BLOCK + wave;  // 0..511
    const int nb    = ntile * 16;
    const int kb    = blockIdx.y * KCHUNK;

    v8f acc0 = {}, acc1 = {}, acc2 = {}, acc3 = {};

    // Row-base pointers (element offsets fit in 32-bit: max 67M)
    const float* xr0 = x + (0 * 16 + l16) * IN_F + 2 * half;
    const float* xr1 = x + (1 * 16 + l16) * IN_F + 2 * half;
    const float* xr2 = x + (2 * 16 + l16) * IN_F + 2 * half;
    const float* xr3 = x + (3 * 16 + l16) * IN_F + 2 * half;
    const float* wr  = w      + (size_t)(nb + l16) * IN_F + 2 * half;
    const float* lr  = logits + (size_t)(nb + l16) * IN_F;   // group base (k%4==0)

    #pragma unroll 2
    for (int k = kb; k < kb + KCHUNK; k += 4) {
        // ---- B fragment: weight masked by hard 2:4 top-k of logits group ----
        v2f    bfrag = *(const v2f*)(wr + k);
        float4 g     = *(const float4*)(lr + k);   // 16B-aligned group of 4

        // Stable top-2 ranks (ties -> lowest index wins, matches lax.top_k)
        int r0 = (g.y > g.x) + (g.z > g.x) + (g.w > g.x);
        int r1 = (g.x > g.y) + (g.z > g.y) + (g.w > g.y) + (g.x == g.y);
        int r2 = (g.x > g.z) + (g.y > g.z) + (g.w > g.z) + (g.x == g.z) + (g.y == g.z);
        int r3 = (g.x > g.w) + (g.y > g.w) + (g.z > g.w) + (g.x == g.w) + (g.y == g.w) + (g.z == g.w);
        int ra = half ? r2 : r0;   // this lane holds group positions 2*half, 2*half+1
        int rb = half ? r3 : r1;
        bfrag.x = (ra < 2) ? bfrag.x : 0.0f;
        bfrag.y = (rb < 2) ? bfrag.y : 0.0f;

        // ---- A fragments for the 4 token tiles (L2-resident x) ----
        v2f a0 = *(const v2f*)(xr0 + k);
        v2f a1 = *(const v2f*)(xr1 + k);
        v2f a2 = *(const v2f*)(xr2 + k);
        v2f a3 = *(const v2f*)(xr3 + k);

        acc0 = __builtin_amdgcn_wmma_f32_16x16x4_f32(false, a0, false, bfrag,
                                                     (short)0, acc0, false, false);
        acc1 = __builtin_amdgcn_wmma_f32_16x16x4_f32(false, a1, false, bfrag,
                                                     (short)0, acc1, false, false);
        acc2 = __builtin_amdgcn_wmma_f32_16x16x4_f32(false, a2, false, bfrag,
                                                     (short)0, acc2, false, false);
        acc3 = __builtin_amdgcn_wmma_f32_16x16x4_f32(false, a3, false, bfrag,
                                                     (short)0, acc3, false, false);
    }

    // ---- Epilogue: K-split partials accumulate atomically onto bias base ----
    // C/D layout: VGPR r, lanes 0-15 -> (M=r, N=lane); lanes 16-31 -> (M=r+8, N=lane-16)
    const int ncol = nb + l16;
    #pragma unroll
    for (int r = 0; r < 8; ++r) {
        const int mrow = r + 8 * half;
        __hip_atomic_fetch_add(&out[(size_t)(0 * 16 + mrow) * OUT_F + ncol], acc0[r],
                               __ATOMIC_RELAXED, __HIP_MEMORY_SCOPE_AGENT);
        __hip_atomic_fetch_add(&out[(size_t)(1 * 16 + mrow) * OUT_F + ncol], acc1[r],
                               __ATOMIC_RELAXED, __HIP_MEMORY_SCOPE_AGENT);
        __hip_atomic_fetch_add(&out[(size_t)(2 * 16 + mrow) * OUT_F + ncol], acc2[r],
                               __ATOMIC_RELAXED, __HIP_MEMORY_SCOPE_AGENT);
        __hip_atomic_fetch_add(&out[(size_t)(3 * 16 + mrow) * OUT_F + ncol], acc3[r],
                               __ATOMIC_RELAXED, __HIP_MEMORY_SCOPE_AGENT);
    }
}

// ---------------------------------------------------------------------------
extern "C" void kernel_launch(void* const* d_in, const int* in_sizes, int n_in,
                              void* d_out, int out_size, void* d_ws, size_t ws_size,
                              hipStream_t stream) {
    const float* x      = (const float*)d_in[0];   // 64 x 8192
    const float* weight = (const float*)d_in[1];   // 8192 x 8192
    const float* bias   = (const float*)d_in[2];   // 8192
    const float* logits = (const float*)d_in[3];   // 8192*8192

    float* out = (float*)d_out;                    // 64 x 8192

    // 1) out = bias (broadcast over rows)
    {
        const int total = TOKENS * OUT_F;          // 524288
        nm_bias_init<<<total / 256, 256, 0, stream>>>(bias, out);
    }

    // 2) masked WMMA GEMM, K split 16 ways, atomic accumulation
    {
        dim3 grid(NTILES / WAVES_PER_BLOCK, KSPLIT);   // (64, 16) blocks
        dim3 block(WAVES_PER_BLOCK * 32);              // 256 threads = 8 waves
        nm_linear_wmma<<<grid, block, 0, stream>>>(x, weight, logits, out);
    }
}